// IRSE_2817498546177
// MI455X (gfx1250) — compile-verified
//
#include <hip/hip_runtime.h>
#include <math.h>

typedef __attribute__((ext_vector_type(2))) float v2f;
typedef __attribute__((ext_vector_type(4))) float v4f;
typedef __attribute__((ext_vector_type(8))) float v8f;

#define PI_F 3.14159265358979323846f

// One K=4 f32 WMMA step: D = A(16x4) * B(4x16) + C
#define WMMA4(a, b, c) \
  __builtin_amdgcn_wmma_f32_16x16x4_f32(false, (a), false, (b), (short)0, (c), false, false)

// Full 16x16 x 16x16 product as 4 chained K=4 WMMAs.
__device__ __forceinline__ v8f mm16(const v2f A[4], const v2f B[4]) {
  v8f c = {};
  c = WMMA4(A[0], B[0], c);
  c = WMMA4(A[1], B[1], c);
  c = WMMA4(A[2], B[2], c);
  c = WMMA4(A[3], B[3], c);
  return c;
}

// Build per-lane A-operand registers for Dblk = diag(D8, D8), D8 = DCT-8 basis.
// A-layout (16x4 f32): lane L holds row M = L%16, K = 4c + 2*(L/16) + j.
__device__ __forceinline__ void make_dblk(int lane, v2f A[4]) {
  const int i = lane & 15;
  const int h = lane >> 4;
  const float sc0 = 0.35355339059327373f;  // sqrt(1/8)
  const float sc1 = 0.5f;                  // sqrt(2/8)
#pragma unroll
  for (int c = 0; c < 4; ++c) {
#pragma unroll
    for (int j = 0; j < 2; ++j) {
      const int kk = 4 * c + 2 * h + j;
      float v = 0.0f;
      if ((i < 8) == (kk < 8)) {
        const int kf = i & 7;   // frequency (row of D)
        const int n  = kk & 7;  // spatial   (col of D)
        const float s = (kf == 0) ? sc0 : sc1;
        v = s * cosf(PI_F * (float)((2 * n + 1) * kf) * (1.0f / 16.0f));
      }
      A[c][j] = v;
    }
  }
}

// Lane holds rows of X: r[v] = X[row = lane%16, col = v + 8*(lane/16)].
// Produce B-layout of X: B[c][j] = X[K = 4c + 2h + j, N = lane%16].
__device__ __forceinline__ void relayout_rows_to_B(float* lds, int lane,
                                                   const float r[8], v2f B[4]) {
  const int i = lane & 15, h = lane >> 4;
#pragma unroll
  for (int v = 0; v < 8; ++v) lds[i * 17 + v + 8 * h] = r[v];
  asm volatile("s_wait_dscnt 0" ::: "memory");
#pragma unroll
  for (int c = 0; c < 4; ++c)
#pragma unroll
    for (int j = 0; j < 2; ++j)
      B[c][j] = lds[(4 * c + 2 * h + j) * 17 + i];
}

// Lane holds C/D layout of X: cd[v] = X[v + 8h, lane%16] (half a column).
// Produce B-layout of X^T: B[c][j] = X^T[4c+2h+j, i] = X[i, 4c+2h+j].
__device__ __forceinline__ void relayout_cols_to_BT(float* lds, int lane,
                                                    const v8f cd, v2f B[4]) {
  const int i = lane & 15, h = lane >> 4;
#pragma unroll
  for (int v = 0; v < 8; ++v) lds[(v + 8 * h) * 17 + i] = cd[v];
  asm volatile("s_wait_dscnt 0" ::: "memory");
#pragma unroll
  for (int c = 0; c < 4; ++c)
#pragma unroll
    for (int j = 0; j < 2; ++j)
      B[c][j] = lds[i * 17 + 4 * c + 2 * h + j];
}

// ---------------------------------------------------------------------------
// Kernel 1: forward DCT on all 16x16 tiles (2x2 DCT blocks each) and per-wave
// deterministic partial sums of each frequency coefficient.
// grid = (128 images, 4 quarters), block = 256 (8 waves), wave -> 8 tiles.
// zpart layout: [b][wave_in_image (32)][64 freq]
// ---------------------------------------------------------------------------
__global__ void __launch_bounds__(256) k_dct_reduce(const float* __restrict__ x,
                                                    float* __restrict__ zpart) {
  __shared__ float lds[8][16 * 17];
  const int b = blockIdx.x, q = blockIdx.y;
  const int lane = threadIdx.x & 31, w = threadIdx.x >> 5;
  const int i = lane & 15, h = lane >> 4;
  float* myl = lds[w];

  v2f A[4];
  make_dblk(lane, A);

  const float* img = x + (size_t)b * 3u * 65536u;  // channel 0
  float zacc[8] = {0.f, 0.f, 0.f, 0.f, 0.f, 0.f, 0.f, 0.f};

  for (int it = 0; it < 8; ++it) {
    const int t = q * 64 + w * 8 + it;
    const int ty = t >> 4, tx = t & 15;
    const float* tp = img + (size_t)(ty * 16) * 256 + tx * 16;

    // Load tile T directly in B-layout: B[c][j] = T[4c + 2h + j, i]
    v2f Bt[4];
#pragma unroll
    for (int c = 0; c < 4; ++c)
#pragma unroll
      for (int j = 0; j < 2; ++j)
        Bt[c][j] = tp[(4 * c + 2 * h + j) * 256 + i];

    const v8f m1 = mm16(A, Bt);            // M1 = Dblk * T      (C/D: cols)
    v2f Bm[4];
    relayout_cols_to_BT(myl, lane, m1, Bm);
    const v8f fT = mm16(A, Bm);            // F^T = Dblk * M1^T  -> fT[v] = F[i, v+8h]

    // Frequency of fT[v]: (k = i%8, l = v). Accumulate.
#pragma unroll
    for (int v = 0; v < 8; ++v) zacc[v] += fT[v];
  }

  // Reduce the 4 lane-copies of each (k,l): lanes {L, L+8, L+16, L+24}.
#pragma unroll
  for (int v = 0; v < 8; ++v) {
    float val = zacc[v];
    val += __shfl_down(val, 16, 32);
    val += __shfl_down(val, 8, 32);
    if (lane < 8) {
      const int widx = q * 8 + w;  // wave index within image: 0..31
      zpart[((size_t)b * 32 + widx) * 64 + lane * 8 + v] = val;
    }
  }
}

// ---------------------------------------------------------------------------
// Kernel 2: per-image SE-MLP. 128 threads, one image each.
// ---------------------------------------------------------------------------
__global__ void k_mlp(const float* __restrict__ zpart,
                      const float* __restrict__ fc1_w, const float* __restrict__ fc1_b,
                      const float* __restrict__ fc2_w, const float* __restrict__ fc2_b,
                      float* __restrict__ s) {
  const int b = blockIdx.x * blockDim.x + threadIdx.x;
  if (b >= 128) return;

  float z[64];
#pragma unroll 4
  for (int f = 0; f < 64; ++f) {
    float acc = 0.0f;
    for (int widx = 0; widx < 32; ++widx)
      acc += zpart[((size_t)b * 32 + widx) * 64 + f];
    z[f] = acc * (1.0f / 1024.0f);  // mean over 32x32 blocks
  }

  float hid[32];
  for (int o = 0; o < 32; ++o) {
    float acc = fc1_b[o];
    for (int f = 0; f < 64; ++f) acc += z[f] * fc1_w[o * 64 + f];
    hid[o] = acc > 0.0f ? acc : 0.0f;  // relu
  }
  for (int j = 0; j < 64; ++j) {
    float acc = fc2_b[j];
    for (int o = 0; o < 32; ++o) acc += hid[o] * fc2_w[j * 32 + o];
    s[b * 64 + j] = 1.0f / (1.0f + expf(-acc));  // sigmoid
  }
}

// ---------------------------------------------------------------------------
// Kernel 3: recompute DCT (x re-read hits L2), scale by s, apply D*(.)*D^T
// again per the reference einsum, write the 256x256 result to all 3 channels.
// ---------------------------------------------------------------------------
__global__ void __launch_bounds__(256) k_apply(const float* __restrict__ x,
                                               const float* __restrict__ s,
                                               float* __restrict__ out) {
  __shared__ float lds[8][16 * 17];
  __shared__ float ssh[64];
  const int b = blockIdx.x, q = blockIdx.y;
  const int lane = threadIdx.x & 31, w = threadIdx.x >> 5;
  const int i = lane & 15, h = lane >> 4;
  float* myl = lds[w];

  if (threadIdx.x < 64) ssh[threadIdx.x] = s[b * 64 + threadIdx.x];
  __syncthreads();

  v2f A[4];
  make_dblk(lane, A);

  const float* img = x + (size_t)b * 3u * 65536u;

  for (int it = 0; it < 8; ++it) {
    const int t = q * 64 + w * 8 + it;
    const int ty = t >> 4, tx = t & 15;
    const float* tp = img + (size_t)(ty * 16) * 256 + tx * 16;

    v2f Bt[4];
#pragma unroll
    for (int c = 0; c < 4; ++c)
#pragma unroll
      for (int j = 0; j < 2; ++j)
        Bt[c][j] = tp[(4 * c + 2 * h + j) * 256 + i];

    const v8f m1 = mm16(A, Bt);             // Dblk * T
    v2f Bm[4];
    relayout_cols_to_BT(myl, lane, m1, Bm);
    const v8f fT = mm16(A, Bm);             // fT[v] = F[i, v+8h]

    // Scale: freq (k = i%8, l = v) -> s[8k + l]
    float g[8];
#pragma unroll
    for (int v = 0; v < 8; ++v) g[v] = fT[v] * ssh[8 * (i & 7) + v];

    // H = Dblk * G : need B-layout of G; we hold G rows per lane.
    v2f Bg[4];
    relayout_rows_to_B(myl, lane, g, Bg);
    const v8f hcd = mm16(A, Bg);            // H in C/D layout (cols per lane)

    // out^T = Dblk * H^T
    v2f Bh[4];
    relayout_cols_to_BT(myl, lane, hcd, Bh);
    const v8f ocd = mm16(A, Bh);            // ocd[v] = out[i, v+8h]

    // Write row (ty*16 + i), cols tx*16 + 8h .. +7, replicated to 3 channels.
    const v4f lo = {ocd[0], ocd[1], ocd[2], ocd[3]};
    const v4f hi = {ocd[4], ocd[5], ocd[6], ocd[7]};
    float* po = out + (size_t)b * 3u * 65536u +
                (size_t)(ty * 16 + i) * 256 + tx * 16 + 8 * h;
#pragma unroll
    for (int ch = 0; ch < 3; ++ch) {
      *(v4f*)(po + (size_t)ch * 65536) = lo;
      *(v4f*)(po + (size_t)ch * 65536 + 4) = hi;
    }
  }
}

extern "C" void kernel_launch(void* const* d_in, const int* in_sizes, int n_in,
                              void* d_out, int out_size, void* d_ws, size_t ws_size,
                              hipStream_t stream) {
  const float* x     = (const float*)d_in[0];
  const float* fc1_w = (const float*)d_in[1];
  const float* fc1_b = (const float*)d_in[2];
  const float* fc2_w = (const float*)d_in[3];
  const float* fc2_b = (const float*)d_in[4];
  float* out = (float*)d_out;

  float* zpart = (float*)d_ws;               // 128 * 32 * 64 floats = 1 MiB
  float* sgate = zpart + 128 * 32 * 64;      // 128 * 64 floats

  dim3 grid(128, 4), blk(256);
  k_dct_reduce<<<grid, blk, 0, stream>>>(x, zpart);
  k_mlp<<<1, 128, 0, stream>>>(zpart, fc1_w, fc1_b, fc2_w, fc2_b, sgate);
  k_apply<<<grid, blk, 0, stream>>>(x, sgate, out);
}